// Net_80968723464705
// MI455X (gfx1250) — compile-verified
//
#include <hip/hip_runtime.h>

typedef __attribute__((ext_vector_type(2))) float v2f;
typedef __attribute__((ext_vector_type(8))) float v8f;

#define IN_F 64
#define OUT_F 64
#define NCLS 16

// ---------------- zero fill (float4 granularity; regions are 256B padded) ----
__global__ void k_zero(float4* __restrict__ p, long n4) {
  long i = (long)blockIdx.x * blockDim.x + threadIdx.x;
  if (i < n4) p[i] = make_float4(0.f, 0.f, 0.f, 0.f);
}

// ---------------- per-edge degree counts --------------------------------------
__global__ void k_degree(const int* __restrict__ src, const int* __restrict__ dst,
                         int* deg_s, int* deg_d, int E) {
  int e = blockIdx.x * blockDim.x + threadIdx.x;
  if (e < E) {
    atomicAdd(&deg_s[src[e]], 1);
    atomicAdd(&deg_d[dst[e]], 1);
  }
}

// ---------------- norm = deg>0 ? deg^{-1/2} : 1 -------------------------------
__global__ void k_norm(const int* __restrict__ deg_s, const int* __restrict__ deg_d,
                       float* ns, float* nd, int N) {
  int i = blockIdx.x * blockDim.x + threadIdx.x;
  if (i < N) {
    int a = deg_s[i]; ns[i] = (a > 0) ? rsqrtf((float)a) : 1.0f;
    int b = deg_d[i]; nd[i] = (b > 0) ? rsqrtf((float)b) : 1.0f;
  }
}

// ---------------- layer-1 scatter: agg[dst] += x[src]*ns[src], 64 feats -------
// 16 lanes per edge, one float4 each -> fully coalesced 256B row reads.
__global__ void k_scatter64(const int* __restrict__ src, const int* __restrict__ dst,
                            const float* __restrict__ x, const float* __restrict__ ns,
                            float* agg, int E) {
  long t = (long)blockIdx.x * blockDim.x + threadIdx.x;
  int e = (int)(t >> 4);
  if (e >= E) return;
  int f = ((int)t & 15) << 2;
  int s = src[e], d = dst[e];
  float w = ns[s];
  float4 v = *(const float4*)(x + (long)s * IN_F + f);
  float* o = agg + (long)d * IN_F + f;
  atomicAdd(o + 0, v.x * w);
  atomicAdd(o + 1, v.y * w);
  atomicAdd(o + 2, v.z * w);
  atomicAdd(o + 3, v.w * w);
}

// ---------------- layer-2 scatter: agg[dst] += z[src]*ns[src], 16 feats -------
__global__ void k_scatter16(const int* __restrict__ src, const int* __restrict__ dst,
                            const float* __restrict__ z, const float* __restrict__ ns,
                            float* agg, int E) {
  long t = (long)blockIdx.x * blockDim.x + threadIdx.x;
  int e = (int)(t >> 2);
  if (e >= E) return;
  int f = ((int)t & 3) << 2;
  int s = src[e], d = dst[e];
  float w = ns[s];
  float4 v = *(const float4*)(z + (long)s * NCLS + f);
  float* o = agg + (long)d * NCLS + f;
  atomicAdd(o + 0, v.x * w);
  atomicAdd(o + 1, v.y * w);
  atomicAdd(o + 2, v.z * w);
  atomicAdd(o + 3, v.w * w);
}

// ---------------- GEMM1: y = relu((A * nd[:,None]) @ W + b), W is 64x64 -------
// One wave per 16-row tile; 4 accumulator tiles (N=64); K stepped by 4 via
// V_WMMA_F32_16X16X4_F32. W staged in LDS (16 KB).
__global__ void __launch_bounds__(256)
k_gemm_relu(const float* __restrict__ A, const float* __restrict__ nd,
            const float* __restrict__ W, const float* __restrict__ bias,
            float* __restrict__ out, int nrows) {
  __shared__ float sW[IN_F * OUT_F];
  for (int i = threadIdx.x; i < IN_F * OUT_F; i += blockDim.x) sW[i] = W[i];
  __syncthreads();

  int wave = (int)((blockIdx.x * blockDim.x + threadIdx.x) >> 5);
  int lane = threadIdx.x & 31;
  int m0 = wave * 16;
  if (m0 >= nrows) return;            // wave-uniform: EXEC stays all-ones

  int lane16 = lane & 15;
  int half   = lane >> 4;             // A/B frags: lanes 0-15 hold K 0,1; 16-31 hold K 2,3
  int arow   = m0 + lane16;
  float scale = nd[arow];
  const float* ap = A + (long)arow * IN_F + half * 2;

  v8f acc0 = {}, acc1 = {}, acc2 = {}, acc3 = {};
#pragma unroll
  for (int k0 = 0; k0 < IN_F; k0 += 4) {
    float2 av = *(const float2*)(ap + k0);
    v2f a = {av.x * scale, av.y * scale};
    int kk = k0 + half * 2;
    const float* w0 = sW + kk * OUT_F + lane16;
    const float* w1 = w0 + OUT_F;
    v2f b0 = {w0[0],  w1[0]};
    v2f b1 = {w0[16], w1[16]};
    v2f b2 = {w0[32], w1[32]};
    v2f b3 = {w0[48], w1[48]};
    acc0 = __builtin_amdgcn_wmma_f32_16x16x4_f32(false, a, false, b0, (short)0, acc0, false, false);
    acc1 = __builtin_amdgcn_wmma_f32_16x16x4_f32(false, a, false, b1, (short)0, acc1, false, false);
    acc2 = __builtin_amdgcn_wmma_f32_16x16x4_f32(false, a, false, b2, (short)0, acc2, false, false);
    acc3 = __builtin_amdgcn_wmma_f32_16x16x4_f32(false, a, false, b3, (short)0, acc3, false, false);
  }

  // C/D layout: lane 0-15 vgpr r -> M=r ; lane 16-31 vgpr r -> M=8+r ; N=lane16
  int rbase = m0 + half * 8;
  float bi0 = bias[lane16], bi1 = bias[16 + lane16];
  float bi2 = bias[32 + lane16], bi3 = bias[48 + lane16];
#pragma unroll
  for (int r = 0; r < 8; ++r) {
    float* orow = out + (long)(rbase + r) * OUT_F + lane16;
    orow[0]  = fmaxf(acc0[r] + bi0, 0.f);
    orow[16] = fmaxf(acc1[r] + bi1, 0.f);
    orow[32] = fmaxf(acc2[r] + bi2, 0.f);
    orow[48] = fmaxf(acc3[r] + bi3, 0.f);
  }
}

// ---------------- GEMM2: z = A @ W2 (64x16), no bias/relu (aggregated later) --
__global__ void __launch_bounds__(256)
k_gemm16(const float* __restrict__ A, const float* __restrict__ W,
         float* __restrict__ out, int nrows) {
  __shared__ float sW[OUT_F * NCLS];
  for (int i = threadIdx.x; i < OUT_F * NCLS; i += blockDim.x) sW[i] = W[i];
  __syncthreads();

  int wave = (int)((blockIdx.x * blockDim.x + threadIdx.x) >> 5);
  int lane = threadIdx.x & 31;
  int m0 = wave * 16;
  if (m0 >= nrows) return;

  int lane16 = lane & 15;
  int half   = lane >> 4;
  const float* ap = A + (long)(m0 + lane16) * OUT_F + half * 2;

  v8f acc = {};
#pragma unroll
  for (int k0 = 0; k0 < OUT_F; k0 += 4) {
    float2 av = *(const float2*)(ap + k0);
    v2f a = {av.x, av.y};
    int kk = k0 + half * 2;
    v2f b = {sW[kk * NCLS + lane16], sW[(kk + 1) * NCLS + lane16]};
    acc = __builtin_amdgcn_wmma_f32_16x16x4_f32(false, a, false, b, (short)0, acc, false, false);
  }
  int rbase = m0 + half * 8;
#pragma unroll
  for (int r = 0; r < 8; ++r)
    out[(long)(rbase + r) * NCLS + lane16] = acc[r];
}

// ---------------- finalize: out = agg2 * nd[:,None] + b2 ----------------------
__global__ void k_finalize(const float* __restrict__ agg, const float* __restrict__ nd,
                           const float* __restrict__ bias, float* __restrict__ out, int N) {
  int i = blockIdx.x * blockDim.x + threadIdx.x;
  if (i < N * NCLS) {
    int n = i >> 4, f = i & 15;
    out[i] = agg[i] * nd[n] + bias[f];
  }
}

extern "C" void kernel_launch(void* const* d_in, const int* in_sizes, int n_in,
                              void* d_out, int out_size, void* d_ws, size_t ws_size,
                              hipStream_t stream) {
  const float* x  = (const float*)d_in[0];
  const int*   src = (const int*)d_in[1];
  const int*   dst = (const int*)d_in[2];
  const float* W1 = (const float*)d_in[3];
  const float* b1 = (const float*)d_in[4];
  const float* W2 = (const float*)d_in[5];
  const float* b2 = (const float*)d_in[6];
  float* out = (float*)d_out;

  const int N = in_sizes[0] / IN_F;   // 100000
  const int E = in_sizes[1];          // 1600000

  // carve workspace (each region 256B aligned)
  char* ws = (char*)d_ws;
  size_t off = 0;
  auto carve = [&](size_t bytes) -> void* {
    void* p = (void*)(ws + off);
    off += (bytes + 255) & ~(size_t)255;
    return p;
  };
  int*   deg_s = (int*)carve((size_t)N * 4);
  int*   deg_d = (int*)carve((size_t)N * 4);
  float* ns    = (float*)carve((size_t)N * 4);
  float* nd    = (float*)carve((size_t)N * 4);
  float* agg1  = (float*)carve((size_t)N * IN_F * 4);
  float* y1    = (float*)carve((size_t)N * OUT_F * 4);
  float* z     = (float*)carve((size_t)N * NCLS * 4);
  float* agg2  = (float*)carve((size_t)N * NCLS * 4);

  const int B = 256;
  auto cdiv = [](long a, long b) { return (int)((a + b - 1) / b); };

  // zero accumulation buffers every call (harness does not re-poison)
  long n4;
  n4 = ((long)N * 4 + 15) / 16;
  k_zero<<<cdiv(n4, B), B, 0, stream>>>((float4*)deg_s, n4);
  k_zero<<<cdiv(n4, B), B, 0, stream>>>((float4*)deg_d, n4);
  n4 = (long)N * IN_F / 4;
  k_zero<<<cdiv(n4, B), B, 0, stream>>>((float4*)agg1, n4);
  n4 = (long)N * NCLS / 4;
  k_zero<<<cdiv(n4, B), B, 0, stream>>>((float4*)agg2, n4);

  // degrees + norms
  k_degree<<<cdiv(E, B), B, 0, stream>>>(src, dst, deg_s, deg_d, E);
  k_norm<<<cdiv(N, B), B, 0, stream>>>(deg_s, deg_d, ns, nd, N);

  // layer 1: aggregate (norm_src folded) -> WMMA gemm (+norm_dst, bias, relu)
  k_scatter64<<<cdiv((long)E * 16, B), B, 0, stream>>>(src, dst, x, ns, agg1, E);
  int waves = cdiv(N, 16);
  k_gemm_relu<<<cdiv((long)waves * 32, B), B, 0, stream>>>(agg1, nd, W1, b1, y1, N);

  // layer 2: matmul hoisted before aggregation (linearity) -> 4x less edge traffic
  k_gemm16<<<cdiv((long)waves * 32, B), B, 0, stream>>>(y1, W2, z, N);
  k_scatter16<<<cdiv((long)E * 4, B), B, 0, stream>>>(src, dst, z, ns, agg2, E);
  k_finalize<<<cdiv((long)N * NCLS, B), B, 0, stream>>>(agg2, nd, b2, out, N);
}